// MapSampler_56985626083915
// MI455X (gfx1250) — compile-verified
//
#include <hip/hip_runtime.h>
#include <stdint.h>

typedef __attribute__((ext_vector_type(2))) float v2f;
typedef __attribute__((ext_vector_type(8))) float v8f;

#define HH   512
#define WW   512
#define HWPIX (HH * WW)        // 262144 pixels per image
#define BLKE 4096              // elements per CDF block (1 workgroup in kernel C)
#define NB   (HWPIX / BLKE)    // 64 blocks per image
#define PT   8192              // samples per image
#define EPS  1e-12f

// ---------------------------------------------------------------- helpers
__device__ __forceinline__ float rlane(float v, int l) {
  return __uint_as_float(__builtin_amdgcn_readlane(__float_as_uint(v), l));
}

// stateless counter-based RNG (splitmix-style), deterministic per (b, s, salt)
__device__ __forceinline__ uint32_t mix3(uint32_t a, uint32_t b, uint32_t c) {
  uint32_t x = a * 0x9E3779B1u + b * 0x85EBCA77u + c * 0xC2B2AE3Du;
  x ^= x >> 16; x *= 0x7FEB352Du;
  x ^= x >> 15; x *= 0x846CA68Bu;
  x ^= x >> 16;
  return x;
}
__device__ __forceinline__ float u01(uint32_t h) {
  return (float)(h >> 8) * (1.0f / 16777216.0f);   // [0,1)
}

// ------------------------------------------------- Kernel A: block sums
// grid = B*NB workgroups of 256 threads; each block sums 4096 weights (+eps)
__global__ void k_block_sums(const float* __restrict__ dmap, float* __restrict__ bs) {
  const int gb = blockIdx.x;                 // b*NB + blk
  const float4* p4 = (const float4*)(dmap + (size_t)gb * BLKE);
  const int t = threadIdx.x;
  float s = 0.0f;
#pragma unroll
  for (int k = 0; k < 4; ++k) {
    float4 v = p4[k * 256 + t];              // coalesced 16B per lane
    s += (v.x + EPS) + (v.y + EPS) + (v.z + EPS) + (v.w + EPS);
  }
  __shared__ float red[256];
  red[t] = s;
  __syncthreads();
#pragma unroll
  for (int o = 128; o > 0; o >>= 1) {
    if (t < o) red[t] += red[t + o];
    __syncthreads();
  }
  if (t == 0) bs[gb] = red[0];
}

// ------------------------------------- Kernel B: scan block sums per image
// grid = B, tiny serial exclusive scan of 64 values (+ total at [64])
__global__ void k_scan_blocks(const float* __restrict__ bs, float* __restrict__ bc) {
  const int b = blockIdx.x;
  if (threadIdx.x == 0) {
    float run = 0.0f;
    const float* s = bs + (size_t)b * NB;
    float* d = bc + (size_t)b * (NB + 1);
#pragma unroll 8
    for (int i = 0; i < NB; ++i) { d[i] = run; run += s[i]; }
    d[NB] = run;                              // per-image total weight
  }
}

// --------------------- Kernel C: full inclusive CDF via WMMA f32 tile scans
// One 256-element tile = 16x16 row-major matrix X.  Within-row inclusive
// prefix = X * U (U upper-triangular ones) as 4 chained V_WMMA_F32_16X16X4_F32.
// Cross-row offsets fixed up with v_readlane (uniform SGPR adds).
__device__ __forceinline__ v8f scan_tile(const float* __restrict__ tbase,
                                         const v2f* __restrict__ bmat,
                                         int m, int hl, float& total) {
  v8f c = {};
#pragma unroll
  for (int k = 0; k < 4; ++k) {
    // A-operand layout (f32 16x4): lane m holds row m; lanes 0-15 carry
    // cols (4k,4k+1), lanes 16-31 carry cols (4k+2,4k+3).
    const float2 raw = *(const float2*)(tbase + m * 16 + 4 * k + 2 * hl);
    v2f a; a.x = raw.x + EPS; a.y = raw.y + EPS;
    c = __builtin_amdgcn_wmma_f32_16x16x4_f32(
        /*neg_a=*/false, a, /*neg_b=*/false, bmat[k],
        /*c_mod=*/(short)0, c, /*reuse_a=*/false, /*reuse_b=*/false);
  }
  // Row totals sit at column 15: lane 15 (rows 0-7) and lane 31 (rows 8-15).
  float rt[16];
#pragma unroll
  for (int r = 0; r < 8; ++r) { rt[r] = rlane(c[r], 15); rt[r + 8] = rlane(c[r], 31); }
  float offs[16]; float run = 0.0f;
#pragma unroll
  for (int i = 0; i < 16; ++i) { offs[i] = run; run += rt[i]; }
#pragma unroll
  for (int r = 0; r < 8; ++r) c[r] += hl ? offs[r + 8] : offs[r];
  total = run;
  return c;
}

// grid = (NB, B), 256 threads = 8 waves; wave scans 2 tiles (512 elems)
__global__ void k_build_cdf(const float* __restrict__ dmap,
                            const float* __restrict__ bc,
                            float* __restrict__ cdf) {
  const int blk = blockIdx.x, b = blockIdx.y;
  const int t = threadIdx.x, lane = t & 31, wv = t >> 5;
  const int m = lane & 15, hl = lane >> 4, n = m;

  const size_t base = (size_t)b * HWPIX + (size_t)blk * BLKE;
  const float* src = dmap + base;
  float*       dst = cdf + base;

  // B-operand: upper-triangular-ones chunks (rows 4k..4k+3), built from ids
  v2f bmat[4];
#pragma unroll
  for (int k = 0; k < 4; ++k) {
    bmat[k].x = (4 * k + 2 * hl     <= n) ? 1.0f : 0.0f;
    bmat[k].y = (4 * k + 2 * hl + 1 <= n) ? 1.0f : 0.0f;
  }

  float t0, t1;
  v8f c0 = scan_tile(src + wv * 512,       bmat, m, hl, t0);
  v8f c1 = scan_tile(src + wv * 512 + 256, bmat, m, hl, t1);

  __shared__ float wtot[8];
  if (lane == 0) wtot[wv] = t0 + t1;
  __syncthreads();

  float add = bc[(size_t)b * (NB + 1) + blk];   // exclusive block base (uniform)
  for (int w = 0; w < wv; ++w) add += wtot[w];
  const float add0 = add, add1 = add + t0;

  // D layout: VGPR r, lanes 0-15 -> row r, lanes 16-31 -> row r+8, col = n
  float* d0 = dst + wv * 512;
  float* d1 = d0 + 256;
#pragma unroll
  for (int r = 0; r < 8; ++r) {
    d0[(r + 8 * hl) * 16 + n] = c0[r] + add0;
    d1[(r + 8 * hl) * 16 + n] = c1[r] + add1;
  }
}

// ----------------------------- Kernel D: categorical draws + uniform prior
// grid = B, 256 threads; coarse search in LDS block-CDF, fine search in the
// L2-resident full CDF (12 steps over a 4096 segment).
__global__ void k_sample(const float* __restrict__ cdf,
                         const float* __restrict__ bc,
                         float* __restrict__ xys,
                         float* __restrict__ prior) {
  const int b = blockIdx.x, t = threadIdx.x;
  __shared__ float sbc[NB + 1];
  for (int i = t; i < NB + 1; i += 256) sbc[i] = bc[(size_t)b * (NB + 1) + i];
  __syncthreads();

  const float total = sbc[NB];
  const float* ic = cdf + (size_t)b * HWPIX;
  const float sx = 2.0f / (float)(WW - 1);
  const float sy = 2.0f / (float)(HH - 1);

  for (int s = t; s < PT; s += 256) {
    const float u = u01(mix3((uint32_t)b, (uint32_t)s, 0x243F6A88u));
    const float tgt = u * total;

    // coarse: largest j with sbc[j] <= tgt  (6 LDS steps)
    int lo = 0, hi = NB - 1;
    while (lo < hi) {
      int mid = (lo + hi + 1) >> 1;
      if (sbc[mid] <= tgt) lo = mid; else hi = mid - 1;
    }
    // fine: smallest i in block with inclusive cdf > tgt (12 L2 steps)
    int l = lo * BLKE, r = l + BLKE - 1;
    while (l < r) {
      int mid = (l + r) >> 1;
      if (ic[mid] > tgt) r = mid; else l = mid + 1;
    }
    const int idx = l;
    const float fx = (float)(idx & (WW - 1)) * sx - 1.0f;
    const float fy = (float)(idx >> 9)       * sy - 1.0f;

    const size_t o = (size_t)b * PT + s;
    xys[o * 2 + 0] = fx;
    xys[o * 2 + 1] = fy;
    prior[o] = u01(mix3((uint32_t)b, (uint32_t)s, 0x85A308D3u));
  }
}

// ---------------------------------------------------------------- launcher
extern "C" void kernel_launch(void* const* d_in, const int* in_sizes, int n_in,
                              void* d_out, int out_size, void* d_ws, size_t ws_size,
                              hipStream_t stream) {
  const float* dmap = (const float*)d_in[0];
  const int Bn = in_sizes[0] / HWPIX;        // 128

  // workspace layout: [cdf: Bn*HWPIX][block sums: Bn*NB][block cdf: Bn*(NB+1)]
  float* cdf = (float*)d_ws;
  float* bs  = cdf + (size_t)Bn * HWPIX;
  float* bc  = bs  + (size_t)Bn * NB;

  float* xys   = (float*)d_out;
  float* prior = xys + (size_t)Bn * PT * 2;

  k_block_sums <<<dim3(Bn * NB), dim3(256), 0, stream>>>(dmap, bs);
  k_scan_blocks<<<dim3(Bn),      dim3(32),  0, stream>>>(bs, bc);
  k_build_cdf  <<<dim3(NB, Bn),  dim3(256), 0, stream>>>(dmap, bc, cdf);
  k_sample     <<<dim3(Bn),      dim3(256), 0, stream>>>(cdf, bc, xys, prior);
}